// tinyBlock_36455682408711
// MI455X (gfx1250) — compile-verified
//
#include <hip/hip_runtime.h>
#include <math.h>

#define DIM 2048
#define HID 8192
#define MAXLEN 8192

typedef __attribute__((ext_vector_type(2))) float v2f;
typedef __attribute__((ext_vector_type(8))) float v8f;

// ---------------------------------------------------------------------------
// Wave-cooperative 16-row GEMV tile using V_WMMA_F32_16X16X4_F32.
//   rowPtr : per-lane pointer to start of weight row (rowBase + (lane&15))
//   xv     : the input vector (uniform)
// Returns D[0][lane] for lanes 0..15 = dot(W[rowBase+lane], xv).
// A-matrix: row 0 = x chunk, rows 1..15 = 0.  B-matrix: 4x16 weight tile.
// Per-lane: VGPR0 = B[2h][n], VGPR1 = B[2h+1][n]  (n = lane&15, h = lane>>4)
// => a single 8-byte load from W[row][k + 2h].
// ---------------------------------------------------------------------------
__device__ __forceinline__ float wmma_gemv_row(const float* __restrict__ rowPtr,
                                               const float* __restrict__ xv,
                                               int K) {
  const int lane = threadIdx.x & 31;
  const int half = lane >> 4;
  const bool isRow0 = (lane & 15) == 0;   // lanes 0 and 16 carry A row 0
  const float* rp = rowPtr + 2 * half;
  const float* xp = xv + 2 * half;
  const v2f z = {0.f, 0.f};
  v8f c = {0.f, 0.f, 0.f, 0.f, 0.f, 0.f, 0.f, 0.f};
#pragma unroll 8
  for (int k = 0; k < K; k += 4) {
    v2f b  = *(const v2f*)(rp + k);       // weight tile column pair
    v2f xa = *(const v2f*)(xp + k);       // broadcast x chunk
    v2f a  = isRow0 ? xa : z;
    c = __builtin_amdgcn_wmma_f32_16x16x4_f32(
        /*neg_a=*/false, a, /*neg_b=*/false, b,
        /*c_mod=*/(short)0, c, /*reuse_a=*/false, /*reuse_b=*/false);
  }
  return c[0];  // VGPR0, lanes 0..15 => D[0][n]
}

// ---------------------------------------------------------------------------
// RMSNorm: out = x / sqrt(mean(x^2)+eps) * w      (single block, 256 threads)
// ---------------------------------------------------------------------------
__global__ void rmsnorm_kernel(const float* __restrict__ x,
                               const float* __restrict__ w,
                               float* __restrict__ out) {
  __shared__ float red[256];
  const int tid = threadIdx.x;
  float s = 0.f;
  for (int i = tid; i < DIM; i += 256) { float v = x[i]; s += v * v; }
  red[tid] = s;
  __syncthreads();
  for (int o = 128; o > 0; o >>= 1) {
    if (tid < o) red[tid] += red[tid + o];
    __syncthreads();
  }
  const float inv = rsqrtf(red[0] * (1.0f / DIM) + 1e-6f);
  for (int i = tid; i < DIM; i += 256) out[i] = x[i] * inv * w[i];
}

// ---------------------------------------------------------------------------
// q/k/v projections: 3*2048 rows, 16 rows per wave => 384 waves (48 blocks)
// ---------------------------------------------------------------------------
__global__ void qkv_kernel(const float* __restrict__ xn,
                           const float* __restrict__ qw, const float* __restrict__ qb,
                           const float* __restrict__ kw, const float* __restrict__ kb,
                           const float* __restrict__ vw, const float* __restrict__ vb,
                           float* __restrict__ q, float* __restrict__ k,
                           float* __restrict__ v) {
  const int wave = (int)((blockIdx.x * blockDim.x + threadIdx.x) >> 5);
  const int mat = wave >> 7;              // 2048/16 = 128 tiles per matrix
  const int rowBase = (wave & 127) * 16;
  const float* W = (mat == 0) ? qw : (mat == 1) ? kw : vw;
  const float* B = (mat == 0) ? qb : (mat == 1) ? kb : vb;
  float* O       = (mat == 0) ? q  : (mat == 1) ? k  : v;
  const int lane = threadIdx.x & 31;
  const int row = rowBase + (lane & 15);
  const float r = wmma_gemv_row(W + (size_t)row * DIM, xn, DIM);
  if (lane < 16) O[row] = r + B[row];
}

// ---------------------------------------------------------------------------
// scores[j] = dot(K_j, q)/sqrt(DIM), j in [0,pos]; row pos uses freshly
// computed k (input cache is NOT mutated). Waves beyond pos exit (uniform).
// ---------------------------------------------------------------------------
__global__ void scores_kernel(const float* __restrict__ cacheK,
                              const float* __restrict__ kNew,
                              const float* __restrict__ q,
                              const int* __restrict__ posPtr,
                              float* __restrict__ scores) {
  const int pos = *posPtr;
  const int wave = (int)((blockIdx.x * blockDim.x + threadIdx.x) >> 5);
  const int rowBase = wave * 16;
  if (rowBase > pos) return;              // wave-uniform early exit
  const int lane = threadIdx.x & 31;
  const int row = rowBase + (lane & 15);
  const float* rowPtr = (row == pos) ? kNew : (cacheK + (size_t)row * DIM);
  const float r = wmma_gemv_row(rowPtr, q, DIM);
  if (lane < 16 && row <= pos)
    scores[row] = r * 0.022097086912079608f;  // 1/sqrt(2048)
}

// ---------------------------------------------------------------------------
// Softmax over scores[0..pos] (single block, 1024 threads)
// ---------------------------------------------------------------------------
__global__ void softmax_kernel(const float* __restrict__ scores,
                               const int* __restrict__ posPtr,
                               float* __restrict__ probs) {
  __shared__ float red[1024];
  const int tid = threadIdx.x;
  const int n = *posPtr + 1;
  float m = -INFINITY;
  for (int i = tid; i < n; i += 1024) m = fmaxf(m, scores[i]);
  red[tid] = m;
  __syncthreads();
  for (int o = 512; o > 0; o >>= 1) {
    if (tid < o) red[tid] = fmaxf(red[tid], red[tid + o]);
    __syncthreads();
  }
  const float mx = red[0];
  __syncthreads();
  float s = 0.f;
  for (int i = tid; i < n; i += 1024) {
    float e = expf(scores[i] - mx);
    probs[i] = e;
    s += e;
  }
  red[tid] = s;
  __syncthreads();
  for (int o = 512; o > 0; o >>= 1) {
    if (tid < o) red[tid] += red[tid + o];
    __syncthreads();
  }
  const float inv = 1.0f / red[0];
  for (int i = tid; i < n; i += 1024) probs[i] *= inv;
}

// ---------------------------------------------------------------------------
// attn[c] = sum_j probs[j]*V_j[c]; row pos uses new v. Column-parallel so
// every V element is read exactly once (8 blocks x 256 threads = 2048 cols).
// ---------------------------------------------------------------------------
__global__ void attnv_kernel(const float* __restrict__ cacheV,
                             const float* __restrict__ vNew,
                             const float* __restrict__ probs,
                             const int* __restrict__ posPtr,
                             float* __restrict__ attn) {
  const int col = (int)(blockIdx.x * blockDim.x + threadIdx.x);
  const int pos = *posPtr;
  float acc = 0.f;
#pragma unroll 4
  for (int j = 0; j < pos; ++j)
    acc += probs[j] * cacheV[(size_t)j * DIM + col];
  acc += probs[pos] * vNew[col];
  attn[col] = acc;
}

// ---------------------------------------------------------------------------
// x1 = x + attn @ o_w.T + o_b   (2048 rows => 16 blocks)
// ---------------------------------------------------------------------------
__global__ void oproj_kernel(const float* __restrict__ ow,
                             const float* __restrict__ ob,
                             const float* __restrict__ attn,
                             const float* __restrict__ xin,
                             float* __restrict__ x1) {
  const int wave = (int)((blockIdx.x * blockDim.x + threadIdx.x) >> 5);
  const int rowBase = wave * 16;
  const int lane = threadIdx.x & 31;
  const int row = rowBase + (lane & 15);
  const float r = wmma_gemv_row(ow + (size_t)row * DIM, attn, DIM);
  if (lane < 16) x1[row] = xin[row] + r + ob[row];
}

// ---------------------------------------------------------------------------
// h = gelu_exact(hn @ fc1_w.T + fc1_b)   (8192 rows => 64 blocks)
// ---------------------------------------------------------------------------
__global__ void fc1_kernel(const float* __restrict__ w,
                           const float* __restrict__ b,
                           const float* __restrict__ hn,
                           float* __restrict__ h) {
  const int wave = (int)((blockIdx.x * blockDim.x + threadIdx.x) >> 5);
  const int rowBase = wave * 16;
  const int lane = threadIdx.x & 31;
  const int row = rowBase + (lane & 15);
  const float r = wmma_gemv_row(w + (size_t)row * DIM, hn, DIM);
  if (lane < 16) {
    const float t = r + b[row];
    h[row] = 0.5f * t * (1.0f + erff(t * 0.70710678118654752f));  // exact GELU
  }
}

// ---------------------------------------------------------------------------
// out = x1 + h @ fc2_w.T + fc2_b   (2048 rows, K=8192 => 16 blocks)
// ---------------------------------------------------------------------------
__global__ void fc2_kernel(const float* __restrict__ w,
                           const float* __restrict__ b,
                           const float* __restrict__ h,
                           const float* __restrict__ x1,
                           float* __restrict__ out) {
  const int wave = (int)((blockIdx.x * blockDim.x + threadIdx.x) >> 5);
  const int rowBase = wave * 16;
  const int lane = threadIdx.x & 31;
  const int row = rowBase + (lane & 15);
  const float r = wmma_gemv_row(w + (size_t)row * HID, h, HID);
  if (lane < 16) out[row] = x1[row] + r + b[row];
}

// ---------------------------------------------------------------------------
// Workspace layout (floats)
// ---------------------------------------------------------------------------
#define WS_XN     0
#define WS_Q      (WS_XN + DIM)
#define WS_K      (WS_Q + DIM)
#define WS_V      (WS_K + DIM)
#define WS_SCORES (WS_V + DIM)
#define WS_PROBS  (WS_SCORES + MAXLEN)
#define WS_ATTN   (WS_PROBS + MAXLEN)
#define WS_X1     (WS_ATTN + DIM)
#define WS_HN     (WS_X1 + DIM)
#define WS_H      (WS_HN + DIM)

extern "C" void kernel_launch(void* const* d_in, const int* in_sizes, int n_in,
                              void* d_out, int out_size, void* d_ws, size_t ws_size,
                              hipStream_t stream) {
  (void)in_sizes; (void)n_in; (void)out_size; (void)ws_size;
  const float* x           = (const float*)d_in[0];
  const float* cacheK      = (const float*)d_in[1];
  const float* cacheV      = (const float*)d_in[2];
  const float* attn_norm_w = (const float*)d_in[3];
  const float* q_w = (const float*)d_in[4];
  const float* q_b = (const float*)d_in[5];
  const float* k_w = (const float*)d_in[6];
  const float* k_b = (const float*)d_in[7];
  const float* v_w = (const float*)d_in[8];
  const float* v_b = (const float*)d_in[9];
  const float* o_w = (const float*)d_in[10];
  const float* o_b = (const float*)d_in[11];
  const float* mlp_norm_w = (const float*)d_in[12];
  const float* fc1_w = (const float*)d_in[13];
  const float* fc1_b = (const float*)d_in[14];
  const float* fc2_w = (const float*)d_in[15];
  const float* fc2_b = (const float*)d_in[16];
  const int*   pos   = (const int*)d_in[17];

  float* ws  = (float*)d_ws;
  float* out = (float*)d_out;

  // 1) xn = rmsnorm(x, attn_norm_w)
  rmsnorm_kernel<<<1, 256, 0, stream>>>(x, attn_norm_w, ws + WS_XN);

  // 2) q/k/v projections: 384 waves @ 8 waves/block
  qkv_kernel<<<48, 256, 0, stream>>>(ws + WS_XN, q_w, q_b, k_w, k_b, v_w, v_b,
                                     ws + WS_Q, ws + WS_K, ws + WS_V);

  // 3) attention scores over cache rows [0..pos] (row pos = new k)
  scores_kernel<<<MAXLEN / 16 / 8, 256, 0, stream>>>(
      cacheK, ws + WS_K, ws + WS_Q, pos, ws + WS_SCORES);

  // 4) softmax
  softmax_kernel<<<1, 1024, 0, stream>>>(ws + WS_SCORES, pos, ws + WS_PROBS);

  // 5) attn = probs @ V  (row pos = new v)
  attnv_kernel<<<DIM / 256, 256, 0, stream>>>(cacheV, ws + WS_V, ws + WS_PROBS,
                                              pos, ws + WS_ATTN);

  // 6) x1 = x + attn @ o_w.T + o_b
  oproj_kernel<<<16, 256, 0, stream>>>(o_w, o_b, ws + WS_ATTN, x, ws + WS_X1);

  // 7) hn = rmsnorm(x1, mlp_norm_w)
  rmsnorm_kernel<<<1, 256, 0, stream>>>(ws + WS_X1, mlp_norm_w, ws + WS_HN);

  // 8) h = gelu(hn @ fc1_w.T + fc1_b)
  fc1_kernel<<<64, 256, 0, stream>>>(fc1_w, fc1_b, ws + WS_HN, ws + WS_H);

  // 9) out = x1 + h @ fc2_w.T + fc2_b
  fc2_kernel<<<16, 256, 0, stream>>>(fc2_w, fc2_b, ws + WS_H, ws + WS_X1, out);
}